// PPRPowerIteration_74929999446094
// MI455X (gfx1250) — compile-verified
//
#include <hip/hip_runtime.h>
#include <stdint.h>

// ---------------- problem constants ----------------
#define N_NODES 100000
#define N_EDGES 3200000
#define N_FEAT  256
#define ALPHA   0.1f
#define NITER   10

#define CH      1024                       // scan chunk size
#define NCH     ((N_NODES + CH - 1) / CH)  // 98 chunks
#define MAXD    160                        // per-row degree cap for stable sort
#define PF      6                          // prefetch distance (edges)
#define SH_CAP  192                        // staged CSR entries per row (>= any real degree)

typedef float v4f __attribute__((ext_vector_type(4)));

// ---------------- CSR construction ----------------

__global__ void k_zero_counts(int* __restrict__ cnt) {
  int i = blockIdx.x * blockDim.x + threadIdx.x;
  if (i < N_NODES) cnt[i] = 0;
}

__global__ void k_hist(const int* __restrict__ row, int* __restrict__ cnt) {
  int e = blockIdx.x * blockDim.x + threadIdx.x;
  if (e < N_EDGES) atomicAdd(&cnt[row[e]], 1);
}

__global__ void k_chunk_reduce(const int* __restrict__ cnt, int* __restrict__ csum) {
  __shared__ int sh[256];
  const int b = blockIdx.x, t = threadIdx.x;
  int s = 0;
  for (int k = 0; k < CH; k += 256) {
    int i = b * CH + k + t;
    s += (i < N_NODES) ? cnt[i] : 0;
  }
  sh[t] = s;
  __syncthreads();
  for (int off = 128; off > 0; off >>= 1) {
    if (t < off) sh[t] += sh[t + off];
    __syncthreads();
  }
  if (t == 0) csum[b] = sh[0];
}

__global__ void k_scan_chunksums(int* __restrict__ csum, int* __restrict__ rp) {
  if (blockIdx.x == 0 && threadIdx.x == 0) {
    int run = 0;
    for (int i = 0; i < NCH; ++i) { int v = csum[i]; csum[i] = run; run += v; }
    rp[N_NODES] = run;  // == N_EDGES
  }
}

__global__ void k_chunk_scan(const int* __restrict__ cnt, const int* __restrict__ csum,
                             int* __restrict__ rp, int* __restrict__ cur) {
  __shared__ int sh[CH];
  const int b = blockIdx.x, t = threadIdx.x;
  const int i = b * CH + t;
  const int v = (i < N_NODES) ? cnt[i] : 0;
  sh[t] = v;
  __syncthreads();
  for (int off = 1; off < CH; off <<= 1) {   // Hillis-Steele inclusive scan
    int tv = (t >= off) ? sh[t - off] : 0;
    __syncthreads();
    sh[t] += tv;
    __syncthreads();
  }
  const int excl = sh[t] - v + csum[b];
  if (i < N_NODES) { rp[i] = excl; cur[i] = excl; }
}

// Bin edge INDICES into CSR order (positions via atomic cursor; order fixed up next).
__global__ void k_bin(const int* __restrict__ row, int* __restrict__ cur,
                      int* __restrict__ scol /* temporarily holds edge idx */) {
  int e = blockIdx.x * blockDim.x + threadIdx.x;
  if (e < N_EDGES) {
    int p = atomicAdd(&cur[row[e]], 1);
    scol[p] = e;
  }
}

// Deterministic fixup: stable-sort each row's edges by original edge index, then
// materialize columns and (1-alpha)-scaled values. Makes FMA order replay-invariant.
__global__ void k_fixup(const int* __restrict__ rp, const int* __restrict__ col,
                        const float* __restrict__ vals,
                        int* __restrict__ scol, float* __restrict__ sval) {
  int n = blockIdx.x * blockDim.x + threadIdx.x;
  if (n >= N_NODES) return;
  const int start = rp[n], end = rp[n + 1];
  const int deg = end - start;
  if (deg <= MAXD) {
    int idx[MAXD];
    for (int i = 0; i < deg; ++i) idx[i] = scol[start + i];
    for (int i = 1; i < deg; ++i) {          // insertion sort, ascending edge idx
      int key = idx[i], j = i - 1;
      while (j >= 0 && idx[j] > key) { idx[j + 1] = idx[j]; --j; }
      idx[j + 1] = key;
    }
    for (int i = 0; i < deg; ++i) {
      int e = idx[i];
      scol[start + i] = col[e];
      sval[start + i] = (1.0f - ALPHA) * vals[e];
    }
  } else {  // statistically unreachable; still correct values
    for (int i = 0; i < deg; ++i) {
      int e = scol[start + i];
      scol[start + i] = col[e];
      sval[start + i] = (1.0f - ALPHA) * vals[e];
    }
  }
}

// ---------------- SpMM power-iteration step ----------------
// One 64-thread block (2 wave32s) per destination row; thread f owns feats [4f,4f+4).
// Row's CSR segment is DMA-staged to LDS via gfx1250 async global->LDS loads
// (ASYNCcnt) with TH_LOAD_NT so the linear CSR stream never evicts the gather
// state from L2. x gathers are 64 lanes x 16B = 1KB fully-coalesced bursts and
// the x/y state (~205MB) is (near-)L2-resident in the 192MB L2; the alpha*local
// stream is loaded non-temporally for the same reason. global_prefetch_b8 pulls
// the row PF edges ahead of the FMA chain.
__global__ __launch_bounds__(64) void k_spmm(const v4f* __restrict__ xv,
                                             const v4f* __restrict__ localv,
                                             const int* __restrict__ rp,
                                             const int* __restrict__ scol,
                                             const float* __restrict__ sval,
                                             v4f* __restrict__ yv) {
  __shared__ int   sh_col[SH_CAP];
  __shared__ float sh_val[SH_CAP];

  const int n = blockIdx.x;
  const int f = threadIdx.x;  // 0..63
  const int start = rp[n];
  const int end   = rp[n + 1];
  const int deg   = end - start;
  const int staged = (deg < SH_CAP) ? deg : SH_CAP;
  const int nchunk = (staged + 63) >> 6;   // 64 entries staged per chunk (2 waves x 32 lanes)

  // ---- async DMA: global (scol/sval segment) -> LDS; all 64 lanes EXEC-active ----
  {
    unsigned lds_c = (unsigned)(uintptr_t)&sh_col[0] + (unsigned)f * 4u;
    unsigned lds_v = (unsigned)(uintptr_t)&sh_val[0] + (unsigned)f * 4u;
    unsigned goff  = ((unsigned)start + (unsigned)f) * 4u;  // byte offset (GVS mode)
    for (int j = 0; j < nchunk; ++j) {
      asm volatile("global_load_async_to_lds_b32 %0, %1, %2 th:TH_LOAD_NT"
                   :: "v"(lds_c + (unsigned)j * 256u),
                      "v"(goff + (unsigned)j * 256u),
                      "s"(scol)
                   : "memory");
      asm volatile("global_load_async_to_lds_b32 %0, %1, %2 th:TH_LOAD_NT"
                   :: "v"(lds_v + (unsigned)j * 256u),
                      "v"(goff + (unsigned)j * 256u),
                      "s"(sval)
                   : "memory");
    }
    asm volatile("s_wait_asynccnt 0x0" ::: "memory");  // drain this wave's async DMAs
  }
  __syncthreads();  // publish both waves' LDS writes

  // streamed, zero-reuse within iteration -> non-temporal (don't evict x/y from L2)
  const v4f lp = __builtin_nontemporal_load(&localv[n * 64 + f]);
  v4f acc;
  acc.x = ALPHA * lp.x; acc.y = ALPHA * lp.y;
  acc.z = ALPHA * lp.z; acc.w = ALPHA * lp.w;

  int e = 0;
  for (; e + 1 < staged; e += 2) {
    const int   c0 = sh_col[e];
    const int   c1 = sh_col[e + 1];
    const float v0 = sh_val[e];
    const float v1 = sh_val[e + 1];
    if (e + PF < staged) {                  // uniform branch: prefetch row PF edges ahead
      const int cp = sh_col[e + PF];
      __builtin_prefetch((const void*)&xv[cp * 64 + f], 0, 1);  // global_prefetch_b8
    }
    const v4f a0 = xv[c0 * 64 + f];
    const v4f a1 = xv[c1 * 64 + f];
    acc.x = fmaf(v0, a0.x, acc.x); acc.y = fmaf(v0, a0.y, acc.y);
    acc.z = fmaf(v0, a0.z, acc.z); acc.w = fmaf(v0, a0.w, acc.w);
    acc.x = fmaf(v1, a1.x, acc.x); acc.y = fmaf(v1, a1.y, acc.y);
    acc.z = fmaf(v1, a1.z, acc.z); acc.w = fmaf(v1, a1.w, acc.w);
  }
  if (e < staged) {
    const int   c = sh_col[e];
    const float v = sh_val[e];
    const v4f a = xv[c * 64 + f];
    acc.x = fmaf(v, a.x, acc.x); acc.y = fmaf(v, a.y, acc.y);
    acc.z = fmaf(v, a.z, acc.z); acc.w = fmaf(v, a.w, acc.w);
  }
  // tail beyond SH_CAP (statistically unreachable): direct global reads
  for (int k = start + staged; k < end; ++k) {
    const int   c = scol[k];
    const float v = sval[k];
    const v4f a = xv[c * 64 + f];
    acc.x = fmaf(v, a.x, acc.x); acc.y = fmaf(v, a.y, acc.y);
    acc.z = fmaf(v, a.z, acc.z); acc.w = fmaf(v, a.w, acc.w);
  }

  yv[n * 64 + f] = acc;  // default (temporal) store: y is next iteration's gather source
}

// ---------------- host driver ----------------

extern "C" void kernel_launch(void* const* d_in, const int* in_sizes, int n_in,
                              void* d_out, int out_size, void* d_ws, size_t ws_size,
                              hipStream_t stream) {
  (void)in_sizes; (void)n_in; (void)out_size; (void)ws_size;

  const float* local_preds = (const float*)d_in[0];
  const int*   edge_row    = (const int*)d_in[1];
  const int*   edge_col    = (const int*)d_in[2];
  const float* edge_vals   = (const float*)d_in[3];

  // carve workspace (256B-aligned regions)
  char* w = (char*)d_ws;
  auto take = [&](size_t bytes) -> void* {
    void* p = (void*)w;
    w += (bytes + 255) & ~(size_t)255;
    return p;
  };
  int*   cnt   = (int*)take((size_t)N_NODES * 4);
  int*   rp    = (int*)take((size_t)(N_NODES + 1) * 4);
  int*   csum  = (int*)take((size_t)NCH * 4);
  int*   cur   = (int*)take((size_t)N_NODES * 4);
  int*   scol  = (int*)take((size_t)N_EDGES * 4);
  float* sval  = (float*)take((size_t)N_EDGES * 4);
  float* bufA  = (float*)take((size_t)N_NODES * N_FEAT * 4);

  // ---- build deterministic CSR (every call; graph-capture safe, fixed shapes) ----
  k_zero_counts<<<(N_NODES + 255) / 256, 256, 0, stream>>>(cnt);
  k_hist<<<(N_EDGES + 255) / 256, 256, 0, stream>>>(edge_row, cnt);
  k_chunk_reduce<<<NCH, 256, 0, stream>>>(cnt, csum);
  k_scan_chunksums<<<1, 1, 0, stream>>>(csum, rp);
  k_chunk_scan<<<NCH, CH, 0, stream>>>(cnt, csum, rp, cur);
  k_bin<<<(N_EDGES + 255) / 256, 256, 0, stream>>>(edge_row, cur, scol);
  k_fixup<<<(N_NODES + 127) / 128, 128, 0, stream>>>(rp, edge_col, edge_vals, scol, sval);

  // ---- 10 power iterations, ping-pong bufA <-> d_out (even count ends in d_out) ----
  const v4f* localv = (const v4f*)local_preds;
  const v4f* src    = (const v4f*)local_preds;
  v4f* pong[2] = { (v4f*)bufA, (v4f*)d_out };
  for (int it = 0; it < NITER; ++it) {
    v4f* dst = pong[it & 1];
    k_spmm<<<N_NODES, 64, 0, stream>>>(src, localv, rp, scol, sval, dst);
    src = dst;
  }
}